// CSFI_26182120636676
// MI455X (gfx1250) — compile-verified
//
#include <hip/hip_runtime.h>
#include <math.h>

// Problem constants (from reference)
#define BB 4
#define CC 128
#define HH 448
#define WW 448
#define HWSZ (HH*WW)
#define HPAT 56            // 448/8 patch rows
#define WPAT 56            // 448/8 patch cols
#define NPB  (HPAT*WPAT)   // 3136 patches per batch
#define PGRP (NPB/16)      // 196 groups of 16 patches
#define MSTR 65            // padded LDS stride for DCT matrix (bank-conflict-free)

// zigzag(8,8,16) selected coefficient indices as a 64-bit mask:
// {0,1,2,3,4,5, 8,9,10,11, 16,17,18, 24,25, 32}
#define ZZMASK 0x103070F3FULL

__constant__ int c_tx[16] = {0,0,6,0,0,1,1,4,5,1,3,0,0,0,3,2};
__constant__ int c_ty[16] = {0,1,0,5,2,0,2,0,0,6,0,4,6,3,5,2};

typedef __attribute__((ext_vector_type(2))) float v2f;
typedef __attribute__((ext_vector_type(8))) float v8f;

__device__ __forceinline__ float basis7(int pos, int freq) {
    float s = (freq == 0) ? rsqrtf(7.0f) : sqrtf(2.0f / 7.0f);
    return cosf(3.14159265358979f * (float)freq * ((float)pos + 0.5f) / 7.0f) * s;
}

// d8[k][i] per reference _dct1d(8)
__device__ __forceinline__ float dct8(int k, int i) {
    float s = (k == 0) ? 0.3535533906f : 0.5f;  // 1/sqrt(8), 1/sqrt(4)
    return cosf(0.39269908169872414f * (float)k * ((float)i + 0.5f)) * s;
}

__device__ __forceinline__ float fsigmoid(float v) {
    return 1.0f / (1.0f + __expf(-v));           // v_exp_f32 path
}

// ---------------------------------------------------------------------------
// Kernel 1: per (b,c,block-row) partial of  sum_{h,w} x * filt(c, h/64, w/64)
// grid = (512, 7), block = 256.  Deterministic (no atomics, no ws zeroing).
// Fully coalesced float4 streaming (contiguous 512B per wave, 4KB per block iteration).
// ---------------------------------------------------------------------------
__global__ void k1_pool_dct(const float* __restrict__ x, float* __restrict__ y_part) {
    const int plane = blockIdx.x;          // b*128 + c
    const int bh    = blockIdx.y;          // pooled row-block 0..6
    const int c     = plane & (CC - 1);
    const int grp   = c >> 3;              // 128/16 = 8 channels per freq group
    const int t     = threadIdx.x;

    __shared__ float fw[7];
    __shared__ float red[256];
    if (t < 7) fw[t] = basis7(bh, c_tx[grp]) * basis7(t, c_ty[grp]);
    __syncthreads();

    // 64 rows x 448 cols = 7168 float4
    const float4* xp = (const float4*)(x + (size_t)plane * HWSZ + (size_t)bh * 64 * WW);
    float acc = 0.0f;
    for (int i = t; i < 7168; i += 256) {
        float4 v = xp[i];
        int j = (i % 112) >> 4;            // w-block = (col/4)/16
        acc += (v.x + v.y + v.z + v.w) * fw[j];
    }
    red[t] = acc;
    __syncthreads();
    for (int s = 128; s > 0; s >>= 1) {
        if (t < s) red[t] += red[t + s];
        __syncthreads();
    }
    if (t == 0) y_part[plane * 7 + bh] = red[0];
}

// ---------------------------------------------------------------------------
// Kernel 2: fold partials -> y[B,C]; SE MLP 128->8->128; sigmoid;
// store channel gate and weff = gate * wc.  grid=1, block=512.
// ---------------------------------------------------------------------------
__global__ void k2_gate(const float* __restrict__ y_part,
                        const float* __restrict__ w1,   // [8,128]
                        const float* __restrict__ w2,   // [128,8]
                        const float* __restrict__ wc,   // [1,128]
                        float* __restrict__ gate,       // [B*C]
                        float* __restrict__ weff) {     // [B*C]
    __shared__ float y_s[BB * CC];
    __shared__ float z_s[BB * 8];
    const int t = threadIdx.x;             // 0..511 -> (b = t/128, c = t%128)

    {
        float s = 0.0f;
        #pragma unroll
        for (int i = 0; i < 7; ++i) s += y_part[t * 7 + i];
        y_s[t] = s * (1.0f / 4096.0f);     // pooled mean over 64*64
    }
    __syncthreads();
    if (t < BB * 8) {
        const int b = t >> 3, j = t & 7;
        float s = 0.0f;
        for (int c = 0; c < CC; ++c) s += y_s[b * CC + c] * w1[j * CC + c];
        z_s[t] = fmaxf(s, 0.0f);           // relu
    }
    __syncthreads();
    {
        const int b = t >> 7, c = t & (CC - 1);
        float s = 0.0f;
        #pragma unroll
        for (int j = 0; j < 8; ++j) s += z_s[b * 8 + j] * w2[c * 8 + j];
        const float g = fsigmoid(s);
        gate[t] = g;
        weff[t] = g * wc[c];
    }
}

// ---------------------------------------------------------------------------
// Kernel 3: fused spatial attention + output.
// One workgroup = (batch b, group of 16 flattened 8x8 patches).
// Thread map: t -> (image row r = t>>5, quad col cq = t&31) so each wave
// touches one full 128-float image row (512B contiguous per access).
//   Phase A: xs[64x16] = sum_c weff[c] * x           (all 8 waves)
//   Phase B: COEF[64x16] = M_masked[64x64] x XS      (waves 0-3, one 16-row
//            M-tile each; 16 chained V_WMMA_F32_16X16X4_F32 per wave)
//            sgate = sigmoid(wp * COEF)
//   Phase C: out = x * cgate[c] * sgate[pixel]       (all 8 waves, L2-hot re-read)
// grid = 4*196 = 784, block = 256.
// ---------------------------------------------------------------------------
__global__ void k3_spatial(const float* __restrict__ x,
                           const float* __restrict__ gate,
                           const float* __restrict__ weff,
                           const float* __restrict__ wp,
                           float* __restrict__ out) {
    __shared__ float M_s[64 * MSTR];   // masked DCT matrix, padded rows
    __shared__ float xs_s[64 * 17];    // XS[n][p], padded stride
    __shared__ float sg_s[64 * 17];    // spatial gate per pixel/patch
    __shared__ float weff_s[CC];
    __shared__ float cg_s[CC];

    const int t = threadIdx.x;                 // 256 threads
    const int b = blockIdx.x / PGRP;
    const int g = blockIdx.x % PGRP;

    if (t < CC) { weff_s[t] = weff[b * CC + t]; cg_s[t] = gate[b * CC + t]; }

    // Build masked DCT matrix: M[m][n] = zz(m) ? d8[mr][nr]*d8[mc][nc] : 0
    {
        const int m  = t >> 2;
        const int nb = (t & 3) * 16;
        const bool on = (ZZMASK >> m) & 1ULL;
        const int mr = m >> 3, mc = m & 7;
        for (int i = 0; i < 16; ++i) {
            const int n = nb + i;
            float v = 0.0f;
            if (on) v = dct8(mr, n >> 3) * dct8(mc, n & 7);
            M_s[m * MSTR + n] = v;
        }
    }
    __syncthreads();

    // Thread -> pixel quad: wave = image row, lanes sweep 128 columns
    const int r   = t >> 5;                    // image row within patch-row 0..7
    const int cq  = t & 31;                    // quad column 0..31
    const int p   = cq >> 1;                   // patch 0..15
    const int n0  = r * 8 + (cq & 1) * 4;      // pixel index r*8+c, quad start
    const int q   = g * 16 + p;                // flattened patch id
    const int h   = (q / WPAT) * 8 + r;
    const int w   = (q % WPAT) * 8 + (cq & 1) * 4;
    const size_t base = (size_t)b * CC * HWSZ + (size_t)h * WW + (size_t)w;

    // Phase A: channel reduction with weff
    float4 acc = make_float4(0.f, 0.f, 0.f, 0.f);
    for (int ch = 0; ch < CC; ++ch) {
        const float4 v = *(const float4*)(x + base + (size_t)ch * HWSZ);
        const float wv = weff_s[ch];
        acc.x += wv * v.x; acc.y += wv * v.y; acc.z += wv * v.z; acc.w += wv * v.w;
    }
    xs_s[(n0 + 0) * 17 + p] = acc.x;
    xs_s[(n0 + 1) * 17 + p] = acc.y;
    xs_s[(n0 + 2) * 17 + p] = acc.z;
    xs_s[(n0 + 3) * 17 + p] = acc.w;
    __syncthreads();

    // Phase B: waves 0-3 each compute one 16-row M-tile of COEF = M x XS.
    // Branch is wave-uniform => EXEC all-ones within participating waves.
    {
        const int wv = t >> 5;                 // wave id
        if (wv < 4) {
            const int mt = wv;
            const int ln = t & 31;
            const int lh = ln >> 4;            // lane half
            const int ll = ln & 15;
            const float wpv = wp[0];
            v8f dacc = {0.f, 0.f, 0.f, 0.f, 0.f, 0.f, 0.f, 0.f};
            const int m = mt * 16 + ll;        // A row (same for both halves)
            #pragma unroll
            for (int kt = 0; kt < 16; ++kt) {
                const int k0 = kt * 4 + lh * 2;
                v2f av; av.x = M_s[m * MSTR + k0];   av.y = M_s[m * MSTR + k0 + 1];
                v2f bv; bv.x = xs_s[k0 * 17 + ll];   bv.y = xs_s[(k0 + 1) * 17 + ll];
                dacc = __builtin_amdgcn_wmma_f32_16x16x4_f32(
                    false, av, false, bv, (short)0, dacc, false, false);
            }
            #pragma unroll
            for (int i = 0; i < 8; ++i) {
                const int mm = mt * 16 + i + lh * 8;   // C/D row layout
                sg_s[mm * 17 + ll] = fsigmoid(wpv * dacc[i]);
            }
        }
    }
    __syncthreads();

    // Phase C: gated output (same 512B/wave-contiguous pattern, L2-hot)
    float4 sg;
    sg.x = sg_s[(n0 + 0) * 17 + p];
    sg.y = sg_s[(n0 + 1) * 17 + p];
    sg.z = sg_s[(n0 + 2) * 17 + p];
    sg.w = sg_s[(n0 + 3) * 17 + p];
    for (int ch = 0; ch < CC; ++ch) {
        const float4 v = *(const float4*)(x + base + (size_t)ch * HWSZ);
        const float cg = cg_s[ch];
        float4 o;
        o.x = v.x * cg * sg.x; o.y = v.y * cg * sg.y;
        o.z = v.z * cg * sg.z; o.w = v.w * cg * sg.w;
        *(float4*)(out + base + (size_t)ch * HWSZ) = o;
    }
}

// ---------------------------------------------------------------------------
extern "C" void kernel_launch(void* const* d_in, const int* in_sizes, int n_in,
                              void* d_out, int out_size, void* d_ws, size_t ws_size,
                              hipStream_t stream) {
    const float* x  = (const float*)d_in[0];
    const float* w1 = (const float*)d_in[1];
    const float* w2 = (const float*)d_in[2];
    const float* wc = (const float*)d_in[3];
    const float* wp = (const float*)d_in[4];
    float* out = (float*)d_out;

    float* wsf    = (float*)d_ws;
    float* y_part = wsf;                 // 512*7 = 3584 floats
    float* gatep  = wsf + 3584;          // 512 floats
    float* weffp  = wsf + 4096;          // 512 floats

    k1_pool_dct<<<dim3(BB * CC, 7), 256, 0, stream>>>(x, y_part);
    k2_gate<<<1, 512, 0, stream>>>(y_part, w1, w2, wc, gatep, weffp);
    k3_spatial<<<BB * PGRP, 256, 0, stream>>>(x, gatep, weffp, wp, out);
}